// CrossAttn_5763846111578
// MI455X (gfx1250) — compile-verified
//
#include <hip/hip_runtime.h>
#include <math.h>

// Problem constants (match reference setup_inputs exactly)
#define B_   2
#define NR   8192
#define NP   8192
#define CH   256
#define KNN  8

typedef __attribute__((ext_vector_type(2))) float v2f;
typedef __attribute__((ext_vector_type(8))) float v8f;

// ---------------------------------------------------------------------------
// Prep: pack ref points as A-fragments {-2x,-2y,-2z,|r|^2} and pred points
// as B-fragments {x,y,z,1} so that A*B = |r|^2 - 2 p.r  (|p|^2 is constant
// per pred point and cannot change top-k ordering, so it is dropped).
// ---------------------------------------------------------------------------
__global__ void prep_pts_kernel(const float* __restrict__ xyz_ref,
                                const float* __restrict__ xyz_pred,
                                float4* __restrict__ ref4,
                                float4* __restrict__ pred4) {
  int i = blockIdx.x * blockDim.x + threadIdx.x;   // NR == NP
  if (i >= B_ * NR) return;
  float rx = xyz_ref[i*3+0], ry = xyz_ref[i*3+1], rz = xyz_ref[i*3+2];
  ref4[i] = make_float4(-2.f*rx, -2.f*ry, -2.f*rz, rx*rx + ry*ry + rz*rz);
  float px = xyz_pred[i*3+0], py = xyz_pred[i*3+1], pz = xyz_pred[i*3+2];
  pred4[i] = make_float4(px, py, pz, 1.0f);
}

// ---------------------------------------------------------------------------
// KNN: one wave handles 16 pred points (N axis of the WMMA). Loop over ref
// tiles of 32 (two 16-ref WMMAs per iteration): each v_wmma_f32_16x16x4_f32
// gives every lane 8 candidate distances for its pred point. A min-of-16
// prefilter + single wave guard skips the whole insert block when no lane
// can improve its register-resident sorted top-8 (the common case once the
// heap warms up). Lanes l and l^16 cover disjoint ref halves of the same
// pred point and are merged at the end via shuffles.
// ---------------------------------------------------------------------------
__device__ __forceinline__ void top8_insert(float c, int ci, float d[KNN], int id[KNN]) {
  if (c < d[KNN-1]) {
#pragma unroll
    for (int j = KNN-1; j > 0; --j) {
      bool lt  = c < d[j];
      bool ltp = c < d[j-1];
      float nd = ltp ? d[j-1] : c;
      int  nid = ltp ? id[j-1] : ci;
      d[j]  = lt ? nd  : d[j];
      id[j] = lt ? nid : id[j];
    }
    if (c < d[0]) { d[0] = c; id[0] = ci; }
  }
}

__global__ __launch_bounds__(256)
void knn_kernel(const float4* __restrict__ ref4,
                const float4* __restrict__ pred4,
                int* __restrict__ knn_idx) {
  const int lane = threadIdx.x & 31;
  const int wave = threadIdx.x >> 5;
  const int tile = blockIdx.x * 8 + wave;   // 16 pred points per wave
  const int pbase = tile * 16;
  const int b  = pbase / NP;
  const int p0 = pbase - b * NP;
  const int half = lane >> 4;               // 0: K=0,1 / refs m0..7 ; 1: K=2,3 / refs m8..15
  const int l15  = lane & 15;

  // B fragment (pred points, constant over the ref loop)
  float4 pt = pred4[b * NP + p0 + l15];
  v2f bfrag;
  bfrag.x = half ? pt.z : pt.x;
  bfrag.y = half ? 1.0f : pt.y;

  float d[KNN]; int id[KNN];
#pragma unroll
  for (int j = 0; j < KNN; ++j) { d[j] = 3.4e38f; id[j] = 0; }

  const float4* refb = ref4 + (size_t)b * NR;
  for (int rt = 0; rt < NR; rt += 32) {
    __builtin_prefetch(&refb[rt + 256 + l15], 0, 0);   // global_prefetch_b8
    float4 rv0 = refb[rt + l15];
    float4 rv1 = refb[rt + 16 + l15];
    v2f a0, a1;
    a0.x = half ? rv0.z : rv0.x;
    a0.y = half ? rv0.w : rv0.y;
    a1.x = half ? rv1.z : rv1.x;
    a1.y = half ? rv1.w : rv1.y;
    v8f acc0 = {}, acc1 = {};
    acc0 = __builtin_amdgcn_wmma_f32_16x16x4_f32(false, a0, false, bfrag,
                                                 (short)0, acc0, false, false);
    acc1 = __builtin_amdgcn_wmma_f32_16x16x4_f32(false, a1, false, bfrag,
                                                 (short)0, acc1, false, false);
    // min-of-16 prefilter: skip all inserts unless some candidate can enter
    float m = fminf(acc0[0], acc0[1]);
#pragma unroll
    for (int r = 2; r < 8; ++r) m = fminf(m, acc0[r]);
#pragma unroll
    for (int r = 0; r < 8; ++r) m = fminf(m, acc1[r]);
    if (m < d[KNN-1]) {
      const int mbase = rt + half * 8;
#pragma unroll
      for (int r = 0; r < 8; ++r)
        top8_insert(acc0[r], mbase + r, d, id);
#pragma unroll
      for (int r = 0; r < 8; ++r)
        top8_insert(acc1[r], mbase + 16 + r, d, id);
    }
  }

  // merge the two half-lane candidate sets of each pred point
#pragma unroll
  for (int j = 0; j < KNN; ++j) {
    float od = __shfl_xor(d[j], 16, 32);
    int   oi = __shfl_xor(id[j], 16, 32);
    top8_insert(od, oi, d, id);
  }

  if (half == 0) {
    int p = pbase + l15;
#pragma unroll
    for (int j = 0; j < KNN; ++j) knn_idx[p * KNN + j] = id[j];
  }
}

// ---------------------------------------------------------------------------
// Generic fp32 WMMA GEMM:  Out[m,n] = sum_k A[m,k] * W[n,k] + bias[n]
// Block: 256 threads / 8 waves, tile 128(M) x 32(N), K staged via LDS (TK=32).
// Each wave owns a 16x32 slab -> two f32 accumulators, 2 WMMAs per k4 step.
// LDS row stride 36 keeps float2 fragments 8B aligned and banks spread.
// ---------------------------------------------------------------------------
__global__ __launch_bounds__(256)
void gemm_wmma_kernel(const float* __restrict__ A,
                      const float* __restrict__ W,
                      const float* __restrict__ bias,
                      float* __restrict__ Out,
                      int M, int N, int Kc,
                      long aBatchStride, long outBatchStride) {
  __shared__ float As[128][36];
  __shared__ float Ws[32][36];
  const int lane = threadIdx.x & 31;
  const int wave = threadIdx.x >> 5;
  const int l15  = lane & 15;
  const int half = lane >> 4;
  const int rowBase = blockIdx.x * 128;
  const int colBase = blockIdx.y * 32;
  A   += (long)blockIdx.z * aBatchStride;
  Out += (long)blockIdx.z * outBatchStride;

  v8f c0 = {}, c1 = {};
  for (int kc = 0; kc < Kc; kc += 32) {
    // stage A tile: 128x32 floats (1024 float4, 4 per thread), coalesced rows
#pragma unroll
    for (int i = 0; i < 4; ++i) {
      int e = threadIdx.x + i * 256;
      int r = e >> 3;
      int s = e & 7;
      float4 v = *(const float4*)&A[(long)(rowBase + r) * Kc + kc + s * 4];
      *(float4*)&As[r][s * 4] = v;
    }
    // stage W tile: 32x32 floats (1 float4 per thread)
    {
      int n = threadIdx.x >> 3;
      int s = threadIdx.x & 7;
      float4 v = *(const float4*)&W[(long)(colBase + n) * Kc + kc + s * 4];
      *(float4*)&Ws[n][s * 4] = v;
    }
    __syncthreads();
    const int mrow = wave * 16 + l15;
#pragma unroll
    for (int k4 = 0; k4 < 8; ++k4) {
      int kk = k4 * 4 + half * 2;
      v2f a  = *(const v2f*)&As[mrow][kk];
      v2f b0 = *(const v2f*)&Ws[l15][kk];
      v2f b1 = *(const v2f*)&Ws[16 + l15][kk];
      c0 = __builtin_amdgcn_wmma_f32_16x16x4_f32(false, a, false, b0, (short)0, c0, false, false);
      c1 = __builtin_amdgcn_wmma_f32_16x16x4_f32(false, a, false, b1, (short)0, c1, false, false);
    }
    __syncthreads();
  }
  // epilogue (D layout: VGPR r -> M=r for lanes 0-15, M=r+8 for lanes 16-31)
  int orow  = rowBase + wave * 16 + half * 8;
  int ocol0 = colBase + l15;
  float bb0 = bias ? bias[ocol0]      : 0.f;
  float bb1 = bias ? bias[ocol0 + 16] : 0.f;
#pragma unroll
  for (int r = 0; r < 8; ++r) {
    Out[(long)(orow + r) * N + ocol0]      = c0[r] + bb0;
    Out[(long)(orow + r) * N + ocol0 + 16] = c1[r] + bb1;
  }
}

// ---------------------------------------------------------------------------
// Per-point 8-neighbor cross attention: one wave per pred point,
// 8 channels per lane, wave32 shfl_xor reductions for the q.k dots.
// ---------------------------------------------------------------------------
__global__ __launch_bounds__(256)
void attn_kernel(const float* __restrict__ q,      // [B*NP, CH]
                 const float* __restrict__ kfeat,  // [B*NR, CH]
                 const float* __restrict__ vall,   // [B*NR, CH]
                 const int* __restrict__ knn_idx,  // [B*NP, KNN]
                 float* __restrict__ out) {        // [B*NP, CH]
  const int lane  = threadIdx.x & 31;
  const int wave  = threadIdx.x >> 5;
  const int point = blockIdx.x * 8 + wave;
  const int b  = point / NP;
  const int c0 = lane * 8;

  float4 q0 = *(const float4*)&q[(long)point * CH + c0];
  float4 q1 = *(const float4*)&q[(long)point * CH + c0 + 4];

  int idx[KNN];
#pragma unroll
  for (int j = 0; j < KNN; ++j) idx[j] = knn_idx[point * KNN + j];

  float sc[KNN];
#pragma unroll
  for (int j = 0; j < KNN; ++j) {
    const float* kr = &kfeat[((long)b * NR + idx[j]) * CH + c0];
    float4 k0 = *(const float4*)kr;
    float4 k1 = *(const float4*)(kr + 4);
    float p = q0.x*k0.x + q0.y*k0.y + q0.z*k0.z + q0.w*k0.w
            + q1.x*k1.x + q1.y*k1.y + q1.z*k1.z + q1.w*k1.w;
    p += __shfl_xor(p, 16, 32);
    p += __shfl_xor(p,  8, 32);
    p += __shfl_xor(p,  4, 32);
    p += __shfl_xor(p,  2, 32);
    p += __shfl_xor(p,  1, 32);
    sc[j] = p * (1.0f / 16.0f);   // 1/sqrt(256)
  }
  float mx = sc[0];
#pragma unroll
  for (int j = 1; j < KNN; ++j) mx = fmaxf(mx, sc[j]);
  float sum = 0.f;
#pragma unroll
  for (int j = 0; j < KNN; ++j) { sc[j] = __expf(sc[j] - mx); sum += sc[j]; }
  float inv = 1.0f / sum;

  float4 o0 = make_float4(0,0,0,0), o1 = make_float4(0,0,0,0);
#pragma unroll
  for (int j = 0; j < KNN; ++j) {
    float w = sc[j] * inv;
    const float* vr = &vall[((long)b * NR + idx[j]) * CH + c0];
    float4 v0 = *(const float4*)vr;
    float4 v1 = *(const float4*)(vr + 4);
    o0.x += w*v0.x; o0.y += w*v0.y; o0.z += w*v0.z; o0.w += w*v0.w;
    o1.x += w*v1.x; o1.y += w*v1.y; o1.z += w*v1.z; o1.w += w*v1.w;
  }
  *(float4*)&out[(long)point * CH + c0]     = o0;
  *(float4*)&out[(long)point * CH + c0 + 4] = o1;
}

// ---------------------------------------------------------------------------
// Zero-fill the ref rows of the output (they are Wout @ zeros == 0).
// ---------------------------------------------------------------------------
__global__ void zero_ref_rows_kernel(float4* __restrict__ out) {
  long t = (long)blockIdx.x * blockDim.x + threadIdx.x;       // float4 units
  const long perBatch = (long)NR * (2 * CH) / 4;
  if (t >= (long)B_ * perBatch) return;
  long b = t / perBatch;
  long r = t - b * perBatch;
  out[b * ((long)(NR + NP) * (2 * CH) / 4) + r] = make_float4(0, 0, 0, 0);
}

// ---------------------------------------------------------------------------
extern "C" void kernel_launch(void* const* d_in, const int* in_sizes, int n_in,
                              void* d_out, int out_size, void* d_ws, size_t ws_size,
                              hipStream_t stream) {
  const float* xyz_ref  = (const float*)d_in[0];
  const float* xyz_pred = (const float*)d_in[1];
  const float* feat_k   = (const float*)d_in[2];
  const float* feat_q   = (const float*)d_in[3];
  const float* feat_v   = (const float*)d_in[4];
  const float* Wv   = (const float*)d_in[5];
  const float* bv   = (const float*)d_in[6];
  const float* Wo   = (const float*)d_in[7];
  const float* bo   = (const float*)d_in[8];
  const float* Wout = (const float*)d_in[9];
  float* out = (float*)d_out;

  // workspace carve-up (~49.3 MB total)
  char* ws = (char*)d_ws;
  float4* ref4  = (float4*)ws;  ws += (size_t)B_ * NR * 4 * sizeof(float);
  float4* pred4 = (float4*)ws;  ws += (size_t)B_ * NP * 4 * sizeof(float);
  int* knn_idx  = (int*)ws;     ws += (size_t)B_ * NP * KNN * sizeof(int);
  float* v_all  = (float*)ws;   ws += (size_t)B_ * NR * CH * sizeof(float);
  float* attn_o = (float*)ws;   ws += (size_t)B_ * NP * CH * sizeof(float);
  float* out2   = (float*)ws;

  // 1) pack point fragments for the WMMA-based KNN
  prep_pts_kernel<<<(B_ * NR + 255) / 256, 256, 0, stream>>>(xyz_ref, xyz_pred, ref4, pred4);

  // 2) KNN top-8 (128 preds per block: 8 waves x 16 preds)
  knn_kernel<<<B_ * NP / 128, 256, 0, stream>>>(ref4, pred4, knn_idx);

  // 3) v_all = feat_v @ Wv^T + bv    (M = B*NR, N = CH, K = CH)
  gemm_wmma_kernel<<<dim3(B_ * NR / 128, CH / 32, 1), 256, 0, stream>>>(
      feat_v, Wv, bv, v_all, B_ * NR, CH, CH, 0, 0);

  // 4) gather + softmax-attention over the 8 neighbors
  attn_kernel<<<B_ * NP / 8, 256, 0, stream>>>(feat_q, feat_k, v_all, knn_idx, attn_o);

  // 5) out2 = attn @ Wo^T + bo
  gemm_wmma_kernel<<<dim3(B_ * NP / 128, CH / 32, 1), 256, 0, stream>>>(
      attn_o, Wo, bo, out2, B_ * NP, CH, CH, 0, 0);

  // 6) ref rows of the final output are exactly zero
  zero_ref_rows_kernel<<<((long)B_ * NR * 2 * CH / 4 + 255) / 256, 256, 0, stream>>>((float4*)out);

  // 7) pred rows: out[b, NR+p, :] = out2[b,p,:] @ Wout^T   (N = 2*CH, batched via grid.z)
  gemm_wmma_kernel<<<dim3(NP / 128, 2 * CH / 32, B_), 256, 0, stream>>>(
      out2, Wout, nullptr, out + (long)NR * 2 * CH,
      NP, 2 * CH, CH, (long)NP * CH, (long)(NR + NP) * 2 * CH);
}